// TriLinear_11510512353766
// MI455X (gfx1250) — compile-verified
//
#include <hip/hip_runtime.h>
#include <hip/hip_bf16.h>

typedef __attribute__((ext_vector_type(2))) float v2f;
typedef __attribute__((ext_vector_type(8))) float v8f;
typedef __attribute__((ext_vector_type(4))) int   v4i;

#define GLOBAL_AS __attribute__((address_space(1)))
#define LDS_AS    __attribute__((address_space(3)))

#define BATCH 32
#define LSEQ  1024
#define DDIM  768

#if __has_builtin(__builtin_amdgcn_global_load_async_to_lds_b128) && \
    __has_builtin(__builtin_amdgcn_s_wait_asynccnt)
#define USE_ASYNC 1
#else
#define USE_ASYNC 0
#endif

#if USE_ASYNC
__device__ __forceinline__ GLOBAL_AS v4i* as_global_v4i(const float* p) {
    return (GLOBAL_AS v4i*)(v4i*)const_cast<float*>(p);
}
__device__ __forceinline__ LDS_AS v4i* as_lds_v4i(float* p) {
    return (LDS_AS v4i*)(v4i*)p;
}
#endif

// ---------------------------------------------------------------------------
// Kernel 1: row logits. qlog[b,q] = Q[b,q,:]·w1 ; klog[b,k] = K[b,k,:]·w2
// One wave32 per row (same row index used for both Q and K side).
// ---------------------------------------------------------------------------
__global__ __launch_bounds__(256)
void trilinear_logits(const float* __restrict__ Q, const float* __restrict__ K,
                      const float* __restrict__ w1, const float* __restrict__ w2,
                      float* __restrict__ qlog, float* __restrict__ klog) {
    const int wid  = threadIdx.x >> 5;
    const int lane = threadIdx.x & 31;
    const int row  = blockIdx.x * 8 + wid;            // 0 .. 32*1024-1

    const float4* qrow = (const float4*)(Q + (size_t)row * DDIM);
    const float4* krow = (const float4*)(K + (size_t)row * DDIM);
    const float4* w1v4 = (const float4*)w1;
    const float4* w2v4 = (const float4*)w2;

    float accq = 0.0f, acck = 0.0f;
    #pragma unroll
    for (int j = lane; j < DDIM / 4; j += 32) {       // 6 iterations
        float4 qv = qrow[j];
        float4 kv = krow[j];
        float4 av = w1v4[j];
        float4 bv = w2v4[j];
        accq += qv.x * av.x + qv.y * av.y + qv.z * av.z + qv.w * av.w;
        acck += kv.x * bv.x + kv.y * bv.y + kv.z * bv.z + kv.w * bv.w;
    }
    #pragma unroll
    for (int off = 16; off > 0; off >>= 1) {
        accq += __shfl_down(accq, off, 32);
        acck += __shfl_down(acck, off, 32);
    }
    if (lane == 0) {
        qlog[row] = accq;
        klog[row] = acck;
    }
}

// ---------------------------------------------------------------------------
// Kernel 2: per-batch GEMM  out[b] = Q_b * diag(w3) * K_b^T + qlog + klog
// 256 threads = 8 wave32, block tile 128x128, K-chunk 16, double-buffered LDS.
// Each wave: 64x32 region = 4x2 tiles of 16x16, f32 WMMA 16x16x4.
// Staging: GLOBAL_LOAD_ASYNC_TO_LDS_B128 software pipeline when available
// (ASYNCcnt-tracked, in-order completion -> s_wait_asynccnt 4 overlaps the
// next chunk's copies with compute); register-staging fallback otherwise.
// ---------------------------------------------------------------------------
__global__ __launch_bounds__(256)
void trilinear_gemm(const float* __restrict__ Q, const float* __restrict__ K,
                    const float* __restrict__ w3,
                    const float* __restrict__ qlog, const float* __restrict__ klog,
                    float* __restrict__ out) {
    constexpr int BM = 128, BN = 128, KB = 16, PITCH = 20; // pitch pads banks
    __shared__ float sA[2][BM * PITCH];   // 2 * 10240 B
    __shared__ float sB[2][BN * PITCH];   // 2 * 10240 B
#if USE_ASYNC
    __shared__ float sW[DDIM];            // staged w3 (3 KB)
#endif

    const int b       = blockIdx.z;
    const int rowBase = blockIdx.x * BM;  // query rows
    const int colBase = blockIdx.y * BN;  // key rows (output columns)
    const int tid     = threadIdx.x;
    const int lane    = tid & 31;
    const int wid     = tid >> 5;
    const int waveM   = wid >> 2;         // 0..1  -> 64-row slab
    const int waveN   = wid & 3;          // 0..3  -> 32-col slab
    const int m16     = lane & 15;
    const int half    = lane >> 4;        // K/row split per ISA layout

    const float* Qb = Q + (size_t)b * LSEQ * DDIM;
    const float* Kb = K + (size_t)b * LSEQ * DDIM;

    v8f acc[4][2];
    #pragma unroll
    for (int i = 0; i < 4; ++i)
        #pragma unroll
        for (int j = 0; j < 2; ++j)
            acc[i][j] = (v8f){};

    const int nChunks = DDIM / KB;        // 48

#if USE_ASYNC
    // ---- async global -> LDS staging (4 b128 copies per thread per chunk) --
    auto issueChunk = [&](int kk, int buf) {
        #pragma unroll
        for (int t = 0; t < 2; ++t) {
            int i  = tid + t * 256;       // float4 index in [0, 512)
            int r  = i >> 2;
            int c4 = (i & 3) * 4;
            const float* ga = Qb + (size_t)(rowBase + r) * DDIM + kk + c4;
            const float* gb = Kb + (size_t)(colBase + r) * DDIM + kk + c4;
            __builtin_amdgcn_global_load_async_to_lds_b128(
                as_global_v4i(ga), as_lds_v4i(&sA[buf][r * PITCH + c4]), 0, 0);
            __builtin_amdgcn_global_load_async_to_lds_b128(
                as_global_v4i(gb), as_lds_v4i(&sB[buf][r * PITCH + c4]), 0, 0);
        }
    };

    if (tid < DDIM / 4)                   // stage w3 once (192 float4)
        *(float4*)&sW[tid * 4] = *(const float4*)(w3 + tid * 4);
    issueChunk(0, 0);

    for (int c = 0; c < nChunks; ++c) {
        const int buf = c & 1;
        if (c + 1 < nChunks) {
            issueChunk((c + 1) * KB, buf ^ 1);
            __builtin_amdgcn_s_wait_asynccnt(4);   // chunk c landed; c+1 in flight
        } else {
            __builtin_amdgcn_s_wait_asynccnt(0);
        }
        __syncthreads();

        const float* A  = sA[buf];
        const float* Bs = sB[buf];
        v2f wch[4];                        // w3 pairs matching B-frag K layout
        #pragma unroll
        for (int k4i = 0; k4i < 4; ++k4i)
            wch[k4i] = *(const v2f*)&sW[c * KB + k4i * 4 + 2 * half];

        #pragma unroll
        for (int k4i = 0; k4i < 4; ++k4i) {
            const int k4 = k4i * 4;
            v2f afrag[4], bfrag[2];
            #pragma unroll
            for (int tm = 0; tm < 4; ++tm) {
                int row = waveM * 64 + tm * 16 + m16;
                afrag[tm] = *(const v2f*)&A[row * PITCH + k4 + 2 * half];
            }
            #pragma unroll
            for (int tn = 0; tn < 2; ++tn) {
                int col = waveN * 32 + tn * 16 + m16;
                bfrag[tn] = *(const v2f*)&Bs[col * PITCH + k4 + 2 * half] * wch[k4i];
            }
            #pragma unroll
            for (int tm = 0; tm < 4; ++tm)
                #pragma unroll
                for (int tn = 0; tn < 2; ++tn)
                    acc[tm][tn] = __builtin_amdgcn_wmma_f32_16x16x4_f32(
                        false, afrag[tm], false, bfrag[tn],
                        (short)0, acc[tm][tn], false, false);
        }
        __syncthreads();                   // buf reusable by next issue
    }
#else
    // ---- fallback: global -> regs -> LDS (w3 folded during staging) -------
    float4 ra[2], rb[2];
    auto gload = [&](int kk) {
        #pragma unroll
        for (int t = 0; t < 2; ++t) {
            int i  = tid + t * 256;
            int r  = i >> 2;
            int c4 = (i & 3) * 4;
            ra[t] = *(const float4*)(Qb + (size_t)(rowBase + r) * DDIM + kk + c4);
            float4 kv = *(const float4*)(Kb + (size_t)(colBase + r) * DDIM + kk + c4);
            float4 wv = *(const float4*)(w3 + kk + c4);
            rb[t] = make_float4(kv.x * wv.x, kv.y * wv.y, kv.z * wv.z, kv.w * wv.w);
        }
    };
    auto sstore = [&](int buf) {
        #pragma unroll
        for (int t = 0; t < 2; ++t) {
            int i  = tid + t * 256;
            int r  = i >> 2;
            int c4 = (i & 3) * 4;
            *(float4*)&sA[buf][r * PITCH + c4] = ra[t];
            *(float4*)&sB[buf][r * PITCH + c4] = rb[t];
        }
    };

    gload(0);
    sstore(0);
    __syncthreads();

    for (int c = 0; c < nChunks; ++c) {
        const int buf = c & 1;
        if (c + 1 < nChunks) gload((c + 1) * KB);

        const float* A  = sA[buf];
        const float* Bs = sB[buf];
        #pragma unroll
        for (int k4 = 0; k4 < KB; k4 += 4) {
            v2f afrag[4], bfrag[2];
            #pragma unroll
            for (int tm = 0; tm < 4; ++tm) {
                int row = waveM * 64 + tm * 16 + m16;
                afrag[tm] = *(const v2f*)&A[row * PITCH + k4 + 2 * half];
            }
            #pragma unroll
            for (int tn = 0; tn < 2; ++tn) {
                int col = waveN * 32 + tn * 16 + m16;
                bfrag[tn] = *(const v2f*)&Bs[col * PITCH + k4 + 2 * half];
            }
            #pragma unroll
            for (int tm = 0; tm < 4; ++tm)
                #pragma unroll
                for (int tn = 0; tn < 2; ++tn)
                    acc[tm][tn] = __builtin_amdgcn_wmma_f32_16x16x4_f32(
                        false, afrag[tm], false, bfrag[tn],
                        (short)0, acc[tm][tn], false, false);
        }

        if (c + 1 < nChunks) sstore((c + 1) & 1);
        __syncthreads();
    }
#endif

    // ---- epilogue: add rank-1 logit terms, write out ----
    float* outb = out + (size_t)b * LSEQ * LSEQ;
    const float* ql = qlog + b * LSEQ;
    const float* kl = klog + b * LSEQ;
    #pragma unroll
    for (int tn = 0; tn < 2; ++tn) {
        const int col = colBase + waveN * 32 + tn * 16 + m16;
        const float klv = kl[col];
        #pragma unroll
        for (int tm = 0; tm < 4; ++tm) {
            const int rbase = rowBase + waveM * 64 + tm * 16 + 8 * half;
            #pragma unroll
            for (int v = 0; v < 8; ++v) {
                const int row = rbase + v;          // C/D layout: M = v + 8*half
                outb[(size_t)row * LSEQ + col] = acc[tm][tn][v] + ql[row] + klv;
            }
        }
    }
}

extern "C" void kernel_launch(void* const* d_in, const int* in_sizes, int n_in,
                              void* d_out, int out_size, void* d_ws, size_t ws_size,
                              hipStream_t stream) {
    const float* q  = (const float*)d_in[0];
    const float* k  = (const float*)d_in[1];
    const float* w1 = (const float*)d_in[2];
    const float* w2 = (const float*)d_in[3];
    const float* w3 = (const float*)d_in[4];
    float* out  = (float*)d_out;
    float* qlog = (float*)d_ws;                      // 32*1024 floats
    float* klog = qlog + BATCH * LSEQ;               // 32*1024 floats

    // 32*1024 rows, 8 rows (waves) per 256-thread block
    trilinear_logits<<<dim3(BATCH * LSEQ / 8), dim3(256), 0, stream>>>(
        q, k, w1, w2, qlog, klog);

    // batch slowest (z) so a whole batch's 6 MB working set stays hot in L2
    trilinear_gemm<<<dim3(LSEQ / 128, LSEQ / 128, BATCH), dim3(256), 0, stream>>>(
        q, k, w3, qlog, klog, out);
}